// Embedding_25975962206822
// MI455X (gfx1250) — compile-verified
//
#include <hip/hip_runtime.h>
#include <math.h>

// Embedding gather + positional encoding for MI455X (gfx1250).
//
// out[t, d] = W[d, idx[t]] * sqrt(512) + PE[t % 2048, d]
//
// Memory-bound: ~135 MB of HBM traffic (W read once — 103 MB fits in the
// 192 MB L2 and is reused ~10x across tokens — plus 32 MB NT output writes).
// WMMA is deliberately not used: the one-hot GEMM formulation would burn
// >3000x the MACs. CDNA5-specific paths used instead:
//   * global_load_async_to_lds_b32 + s_wait_asynccnt (async index staging)
//   * non-temporal global_store_b128 (preserve W's L2 residency)

#define D_MODEL 512
#define TOKENS_PER_BLOCK 8   // 8 wave32s per 256-thread block, 1 token/wave

typedef float v4f __attribute__((ext_vector_type(4)));

__global__ __launch_bounds__(256) void embed_pe_kernel(
    const int* __restrict__ idx, const float* __restrict__ W,
    float* __restrict__ out, int n_tokens, int vocab, int seq_mask)
{
    __shared__ int s_idx[TOKENS_PER_BLOCK];
    const int tok0 = blockIdx.x * TOKENS_PER_BLOCK;

    // --- Async-stage this block's 8 token indices into LDS (ASYNCcnt path).
    if (threadIdx.x < TOKENS_PER_BLOCK) {
        // Low 32 bits of the generic pointer to a __shared__ object are the
        // LDS byte offset (generic = {aperture_hi, lds_offset}).
        unsigned lds_off = (unsigned)(size_t)(&s_idx[threadIdx.x]);
        const int* gsrc = idx + tok0 + threadIdx.x;
        asm volatile("global_load_async_to_lds_b32 %0, %1, off"
                     :: "v"(lds_off), "v"(gsrc)
                     : "memory");
    }
    asm volatile("s_wait_asynccnt 0x0" ::: "memory");
    __syncthreads();

    const int wave = threadIdx.x >> 5;   // wave32
    const int lane = threadIdx.x & 31;
    const int tok  = tok0 + wave;
    if (tok >= n_tokens) return;

    const int   v = s_idx[wave];                 // vocab index for this token
    const float s = (float)(tok & seq_mask);     // sequence position

    const float kC     = 0.017988946039015984f;  // ln(10000)/512
    const float kScale = 22.62741699796952f;     // sqrt(512)

    const float* wcol = W + v;                   // element (d, v) at wcol[d*vocab]
    float*       orow = out + (size_t)tok * D_MODEL;

#pragma unroll
    for (int it = 0; it < 4; ++it) {
        const int d0 = it * 128 + lane * 4;      // 4 consecutive features/lane

        // Strided column gather: 4 independent global_load_b32 per lane,
        // ~L2-resident after first touch.
        const float w0 = wcol[(size_t)(d0 + 0) * vocab];
        const float w1 = wcol[(size_t)(d0 + 1) * vocab];
        const float w2 = wcol[(size_t)(d0 + 2) * vocab];
        const float w3 = wcol[(size_t)(d0 + 3) * vocab];

        // PE on the fly: pairs (even d -> sin, odd d -> cos) share a frequency.
        const float f0 = __expf(-(float)(d0    ) * kC);
        const float f1 = __expf(-(float)(d0 + 2) * kC);
        const float a0 = s * f0;
        const float a1 = s * f1;

        v4f r;
        r.x = fmaf(w0, kScale, __sinf(a0));
        r.y = fmaf(w1, kScale, __cosf(a0));
        r.z = fmaf(w2, kScale, __sinf(a1));
        r.w = fmaf(w3, kScale, __cosf(a1));

        // Coalesced 512 B/wave non-temporal store (out is never re-read).
        __builtin_nontemporal_store(r, (v4f*)(orow + d0));
    }
}

extern "C" void kernel_launch(void* const* d_in, const int* in_sizes, int n_in,
                              void* d_out, int out_size, void* d_ws, size_t ws_size,
                              hipStream_t stream) {
    (void)n_in; (void)out_size; (void)d_ws; (void)ws_size;

    const int*   idx = (const int*)d_in[0];      // [8, 2048] token ids
    const float* W   = (const float*)d_in[1];    // [512, 50257] row-major
    float*       out = (float*)d_out;            // [8, 2048, 512]

    const int n_tokens = in_sizes[0];            // 16384
    const int vocab    = in_sizes[1] / D_MODEL;  // 50257
    const int seq_mask = 2047;                   // seq_length = 2048 (ref const)

    const int blocks = (n_tokens + TOKENS_PER_BLOCK - 1) / TOKENS_PER_BLOCK;
    embed_pe_kernel<<<blocks, 256, 0, stream>>>(idx, W, out, n_tokens, vocab, seq_mask);
}